// SSMBlock_86569360818303
// MI455X (gfx1250) — compile-verified
//
#include <hip/hip_runtime.h>
#include <hip/hip_bf16.h>
#include <math.h>

// ---------------------------------------------------------------------------
// Mamba SSM block for gfx1250 (MI455X).
// Pipeline: GEMM(in-proj, f32 WMMA) -> depthwise conv+SiLU -> GEMM(x-proj)
//           -> GEMM(dt-proj + bias + softplus) -> selective scan (8 lanes per
//           channel, LDS-staged operands, fused D-skip + SiLU gate)
//           -> GEMM(out-proj).
// All GEMMs use V_WMMA_F32_16X16X4_F32 (full f32 precision matrix path).
// ---------------------------------------------------------------------------

typedef float v2f __attribute__((ext_vector_type(2)));
typedef float v8f __attribute__((ext_vector_type(8)));

#define DM   1024   // d_model
#define DI   2048   // d_inner
#define NS   16     // d_state
#define KC   4      // d_conv
#define DTR  64     // dt_rank
#define LSEQ 2048   // sequence length
#define XDBL_W (DTR + 2*NS)   // 96

// ---------------------------------------------------------------------------
// Generic f32 WMMA GEMM: C[M,N] = A[M,K] @ B[K,N]  (all row-major).
// Each wave computes a (WM*16) x (WN*16) tile, K-loop step 4.
// A-frag (16x4 f32, 2 VGPR): lane = {r = lane&15, h = lane>>4}
//   a[v] = A[r][k0 + 2h + v]
// B-frag (4x16 f32, 2 VGPR): b[v] = B[k0 + 2h + v][lane&15]
// C/D (16x16 f32, 8 VGPR):   c[v] = C[v + 8h][lane&15]
// ---------------------------------------------------------------------------
template<int WM, int WN, bool SOFTPLUS_BIAS>
__global__ __launch_bounds__(256)
void wmma_gemm_f32(const float* __restrict__ A, const float* __restrict__ B,
                   float* __restrict__ C, const float* __restrict__ bias,
                   int M, int N, int K, int lda, int ldb, int ldc)
{
    const int wave = (blockIdx.x * blockDim.x + threadIdx.x) >> 5;
    const int lane = threadIdx.x & 31;
    const int tilesM = M / (WM * 16);
    const int tilesN = N / (WN * 16);
    if (wave >= tilesM * tilesN) return;      // whole-wave uniform exit
    const int tm = wave % tilesM;
    const int tn = wave / tilesM;
    const int m0 = tm * WM * 16;
    const int n0 = tn * WN * 16;
    const int lr = lane & 15;
    const int lh = lane >> 4;

    v8f acc[WM][WN];
#pragma unroll
    for (int i = 0; i < WM; ++i)
#pragma unroll
        for (int j = 0; j < WN; ++j) {
            v8f z = {0.f, 0.f, 0.f, 0.f, 0.f, 0.f, 0.f, 0.f};
            acc[i][j] = z;
        }

    for (int k0 = 0; k0 < K; k0 += 4) {
        v2f a[WM];
#pragma unroll
        for (int i = 0; i < WM; ++i) {
            const float* ap = A + (size_t)(m0 + i * 16 + lr) * lda + k0 + 2 * lh;
            a[i] = *(const v2f*)ap;           // consecutive K -> b64 load
        }
        v2f b[WN];
#pragma unroll
        for (int j = 0; j < WN; ++j) {
            const float* bp = B + (size_t)(k0 + 2 * lh) * ldb + n0 + j * 16 + lr;
            v2f t; t[0] = bp[0]; t[1] = bp[ldb];
            b[j] = t;
        }
#pragma unroll
        for (int i = 0; i < WM; ++i)
#pragma unroll
            for (int j = 0; j < WN; ++j)
                acc[i][j] = __builtin_amdgcn_wmma_f32_16x16x4_f32(
                    false, a[i], false, b[j], (short)0, acc[i][j], false, false);
    }

#pragma unroll
    for (int i = 0; i < WM; ++i) {
#pragma unroll
        for (int j = 0; j < WN; ++j) {
            const int col = n0 + j * 16 + lr;
#pragma unroll
            for (int r = 0; r < 8; ++r) {
                const int row = m0 + i * 16 + r + 8 * lh;
                float v = acc[i][j][r];
                if (SOFTPLUS_BIAS) {
                    v += bias[col];
                    v = (v > 20.f) ? v : __logf(1.f + __expf(v));  // softplus
                }
                C[(size_t)row * ldc + col] = v;
            }
        }
    }
}

// ---------------------------------------------------------------------------
// Causal depthwise conv (k=4) + bias + SiLU.  xc = x_and_res[:, :DI].
// One thread per (l, d); d innermost for coalescing.
// ---------------------------------------------------------------------------
__global__ __launch_bounds__(256)
void conv_silu_kernel(const float* __restrict__ xr, const float* __restrict__ conv_w,
                      const float* __restrict__ conv_b, float* __restrict__ xs)
{
    const int idx = blockIdx.x * 256 + threadIdx.x;   // [0, LSEQ*DI)
    const int l = idx / DI;
    const int d = idx % DI;
    const float4 w = *(const float4*)(conv_w + (size_t)d * KC);
    float acc = conv_b[d];
    const float* xc = xr + d;                          // stride 2*DI, col d
    if (l >= 3) acc += xc[(size_t)(l - 3) * (2 * DI)] * w.x;
    if (l >= 2) acc += xc[(size_t)(l - 2) * (2 * DI)] * w.y;
    if (l >= 1) acc += xc[(size_t)(l - 1) * (2 * DI)] * w.z;
    acc             += xc[(size_t)l       * (2 * DI)] * w.w;
    const float sig = 1.f / (1.f + __expf(-acc));
    xs[idx] = acc * sig;                               // SiLU
}

// ---------------------------------------------------------------------------
// Selective scan v3: 8 lanes per channel, 2 states per lane.
//   thread t -> channel d = t>>3, state group q = t&7 (states 2q, 2q+1).
// Each block covers 32 contiguous channels.  Per 64-step chunk, dt/xs slices
// and B/C rows are cooperatively staged into LDS (24 KB), so the step loop
// reads only LDS (broadcast b32 for dt/xs, b64 for B/C fragments).
// Per-channel y-dot finished with a 3-step shfl_xor butterfly (masks 1/2/4,
// inside each 8-lane group of the wave32).
// Fuses y = (C.h + xs*D) * silu(res); writes final h (float2) to hT.
// 16384 threads -> 64 blocks x 256 -> 512 waves device-wide.
// ---------------------------------------------------------------------------
__global__ __launch_bounds__(256)
void scan_kernel(const float* __restrict__ dt, const float* __restrict__ xdbl,
                 const float* __restrict__ xs, const float* __restrict__ xr,
                 const float* __restrict__ A_log, const float* __restrict__ Dvec,
                 float* __restrict__ y, float* __restrict__ hT)
{
    __shared__ float sDT[64 * 32];                     // 64 steps x 32 channels
    __shared__ float sXS[64 * 32];
    __shared__ float sBC[64 * 32];                     // 64 steps x (B16|C16)

    const int t  = blockIdx.x * 256 + threadIdx.x;
    const int d  = t >> 3;                             // global channel
    const int q  = t & 7;                              // state group (2 states)
    const int ch = threadIdx.x >> 3;                   // channel within block
    const int d0 = blockIdx.x * 32;                    // first channel of block

    float A0 = -__expf(A_log[(size_t)d * NS + 2 * q]);
    float A1 = -__expf(A_log[(size_t)d * NS + 2 * q + 1]);
    float h0 = 0.f, h1 = 0.f;
    const float Dd = Dvec[d];

    for (int c = 0; c < LSEQ / 64; ++c) {
        __syncthreads();
        for (int e = threadIdx.x; e < 64 * 32; e += 256) {
            const int ls = e >> 5, w = e & 31;
            const int l = c * 64 + ls;
            sDT[e] = dt[(size_t)l * DI + d0 + w];
            sXS[e] = xs[(size_t)l * DI + d0 + w];
            sBC[e] = xdbl[(size_t)l * XDBL_W + DTR + w];
        }
        __syncthreads();
        for (int s = 0; s < 64; ++s) {
            const int l = c * 64 + s;
            const float dtv = sDT[s * 32 + ch];
            const float xv  = sXS[s * 32 + ch];
            const float2 Bv = *(const float2*)&sBC[s * 32 + 2 * q];        // b64
            const float2 Cv = *(const float2*)&sBC[s * 32 + 16 + 2 * q];   // b64
            const float dx  = dtv * xv;
            const float dA0 = __expf(dtv * A0);
            const float dA1 = __expf(dtv * A1);
            h0 = dA0 * h0 + dx * Bv.x;
            h1 = dA1 * h1 + dx * Bv.y;
            float yacc = h0 * Cv.x + h1 * Cv.y;
            // butterfly sum across the 8-lane group (wave32 shfl)
            yacc += __shfl_xor(yacc, 1);
            yacc += __shfl_xor(yacc, 2);
            yacc += __shfl_xor(yacc, 4);
            if (q == 0) {
                const float resv = xr[(size_t)l * (2 * DI) + DI + d];
                const float yv   = yacc + xv * Dd;
                const float sig  = 1.f / (1.f + __expf(-resv));
                y[(size_t)l * DI + d] = yv * (resv * sig);   // * silu(res)
            }
        }
    }
    float2 hout = make_float2(h0, h1);
    *(float2*)&hT[(size_t)d * NS + 2 * q] = hout;
}

// ---------------------------------------------------------------------------
extern "C" void kernel_launch(void* const* d_in, const int* in_sizes, int n_in,
                              void* d_out, int out_size, void* d_ws, size_t ws_size,
                              hipStream_t stream) {
    const float* x      = (const float*)d_in[0];
    const float* W_in   = (const float*)d_in[1];
    const float* conv_w = (const float*)d_in[2];
    const float* conv_b = (const float*)d_in[3];
    const float* W_x    = (const float*)d_in[4];
    const float* W_dt   = (const float*)d_in[5];
    const float* b_dt   = (const float*)d_in[6];
    const float* A_log  = (const float*)d_in[7];
    const float* Dv     = (const float*)d_in[8];
    const float* W_out  = (const float*)d_in[9];

    float* out = (float*)d_out;                       // [LSEQ, DM]
    float* hT  = out + (size_t)LSEQ * DM;             // [DI, NS]

    float* ws   = (float*)d_ws;
    float* xr   = ws;                                 // [LSEQ, 2*DI]  x_and_res
    float* xs   = xr   + (size_t)LSEQ * 2 * DI;       // [LSEQ, DI]    silu(conv)
    float* xdbl = xs   + (size_t)LSEQ * DI;           // [LSEQ, 96]
    float* dt   = xdbl + (size_t)LSEQ * XDBL_W;       // [LSEQ, DI]
    float* yb   = dt   + (size_t)LSEQ * DI;           // [LSEQ, DI]

    // GEMM1: x[2048,1024] @ W_in[1024,4096] -> xr.  32x64 tiles -> 2048 waves.
    wmma_gemm_f32<4, 4, false><<<256, 256, 0, stream>>>(
        x, W_in, xr, nullptr, LSEQ, 2 * DI, DM, DM, 2 * DI, 2 * DI);

    // Depthwise conv + SiLU -> xs.
    conv_silu_kernel<<<(LSEQ * DI) / 256, 256, 0, stream>>>(xr, conv_w, conv_b, xs);

    // GEMM2: xs[2048,2048] @ W_x[2048,96] -> xdbl.  tiles 32x3 (64x32 per wave).
    wmma_gemm_f32<4, 2, false><<<12, 256, 0, stream>>>(
        xs, W_x, xdbl, nullptr, LSEQ, XDBL_W, DI, DI, XDBL_W, XDBL_W);

    // GEMM3: dt_r[2048,64] @ W_dt[64,2048] + b_dt, softplus -> dt.  (lda=96 slice)
    wmma_gemm_f32<4, 4, true><<<128, 256, 0, stream>>>(
        xdbl, W_dt, dt, b_dt, LSEQ, DI, DTR, XDBL_W, DI, DI);

    // Selective scan (8 lanes/channel, fused D-skip + SiLU gate) -> yb, hT.
    scan_kernel<<<(DI * 8) / 256, 256, 0, stream>>>(dt, xdbl, xs, xr, A_log, Dv, yb, hT);

    // GEMM4: yb[2048,2048] @ W_out[2048,1024] -> out.  tiles 32x16 -> 512 waves.
    wmma_gemm_f32<4, 4, false><<<64, 256, 0, stream>>>(
        yb, W_out, out, nullptr, LSEQ, DM, DI, DI, DM, DM);
}